// R2E_44641890075193
// MI455X (gfx1250) — compile-verified
//
#include <hip/hip_runtime.h>
#include <hip/hip_bf16.h>
#include <stdint.h>

typedef __attribute__((ext_vector_type(2))) float v2f;
typedef __attribute__((ext_vector_type(8))) float v8f;

#define NEG_SLOPE 0.01f

__device__ __forceinline__ float lrelu(float x) {
    return x > 0.0f ? x : NEG_SLOPE * x;
}

// Order-preserving float <-> uint encoding so atomicMax(u32) implements float max.
__device__ __forceinline__ unsigned fkey(float f) {
    unsigned u = __float_as_uint(f);
    return (u & 0x80000000u) ? ~u : (u | 0x80000000u);
}
__device__ __forceinline__ float fdecode(unsigned k) {
    unsigned u = (k & 0x80000000u) ? (k ^ 0x80000000u) : ~k;
    return __uint_as_float(u);
}
#define NEG_INF_KEY 0x007FFFFFu  // fkey(-inf)

// ---------------------------------------------------------------------------
// Kernel 0: initialize output accumulators + softmax stats.
// ---------------------------------------------------------------------------
__global__ void k_init(float* __restrict__ out, float* __restrict__ sumh,
                       float* __restrict__ sumt, unsigned* __restrict__ maxh,
                       unsigned* __restrict__ maxt, int n_out, int n_seg) {
    int i = blockIdx.x * blockDim.x + threadIdx.x;
    if (i < n_out) out[i] = 0.0f;
    if (i < n_seg) {
        sumh[i] = 0.0f;
        sumt[i] = 0.0f;
        maxh[i] = NEG_INF_KEY;
        maxt[i] = NEG_INF_KEY;
    }
}

// ---------------------------------------------------------------------------
// Kernel 1: fp32 projection via V_WMMA_F32_16X16X4_F32.
//   S[row, n] = dot(X[row, :], W(n)[:])  for n = 0..15
//   W rows 0..7 come from W0, rows 8..15 from W1.
//   (For the relation launch we pass W1 == W0; columns 8..15 of S then hold a
//    duplicate projection that downstream kernels never read — this avoids any
//    per-iteration masking VALU op in front of the WMMAs.)
// One wave32 per 16-row M tile; K = 128 consumed in 32 steps of 4.
// ---------------------------------------------------------------------------
__global__ void k_proj_wmma(const float* __restrict__ X,
                            const float* __restrict__ W0,
                            const float* __restrict__ W1,
                            float* __restrict__ S, int nrows) {
    const int wave = (blockIdx.x * blockDim.x + threadIdx.x) >> 5;
    const int lane = threadIdx.x & 31;
    const int row0 = wave * 16;
    if (row0 >= nrows) return;  // wave-uniform: EXEC stays all-ones for WMMA

    const int m    = lane & 15;        // A-matrix row this lane feeds / B col
    const int kofs = (lane >> 4) << 1; // 0 for lanes 0-15, 2 for lanes 16-31

    const float* __restrict__ arow = X + (size_t)(row0 + m) * 128;
    const float* __restrict__ brow =
        (m < 8) ? (W0 + (size_t)m * 128) : (W1 + (size_t)(m - 8) * 128);

    v8f c = {0.f, 0.f, 0.f, 0.f, 0.f, 0.f, 0.f, 0.f};
#pragma unroll
    for (int k = 0; k < 128; k += 4) {
        v2f a = *(const v2f*)(arow + k + kofs);
        v2f b = *(const v2f*)(brow + k + kofs);
        // 8 args: (neg_a, A, neg_b, B, c_mod, C, reuse_a, reuse_b)
        c = __builtin_amdgcn_wmma_f32_16x16x4_f32(
            /*neg_a=*/false, a, /*neg_b=*/false, b,
            /*c_mod=*/(short)0, c, /*reuse_a=*/false, /*reuse_b=*/false);
    }

    // D layout: VGPR v, lanes 0-15 -> M=v, lanes 16-31 -> M=v+8; N = lane&15.
    const int rbase = row0 + ((lane >> 4) << 3);
    const int col   = lane & 15;
#pragma unroll
    for (int v = 0; v < 8; ++v) {
        S[(size_t)(rbase + v) * 16 + col] = c[v];
    }
}

// ---------------------------------------------------------------------------
// Kernel 2: per (edge, head) atomic max of leaky-relu scores into seg max.
// ---------------------------------------------------------------------------
__global__ void k_edge_max(const int* __restrict__ h_idx,
                           const int* __restrict__ t_idx,
                           const int* __restrict__ rel,
                           const float* __restrict__ Se,
                           const float* __restrict__ Sr,
                           unsigned* __restrict__ maxh,
                           unsigned* __restrict__ maxt, int E) {
    int i = blockIdx.x * blockDim.x + threadIdx.x;
    if (i >= E * 8) return;
    const int e = i >> 3, h = i & 7;
    const int hi = h_idx[e], ti = t_idx[e], r = rel[e];
    const float er = Sr[(size_t)r * 16 + h];
    const float sh = lrelu(Se[(size_t)hi * 16 + h] + er);
    const float st = lrelu(Se[(size_t)ti * 16 + 8 + h] + er);
    atomicMax(&maxh[(size_t)hi * 8 + h], fkey(sh));
    atomicMax(&maxt[(size_t)ti * 8 + h], fkey(st));
}

// ---------------------------------------------------------------------------
// Kernel 3: per (edge, head) exp(s - segmax) accumulated into seg sums.
// ---------------------------------------------------------------------------
__global__ void k_edge_sum(const int* __restrict__ h_idx,
                           const int* __restrict__ t_idx,
                           const int* __restrict__ rel,
                           const float* __restrict__ Se,
                           const float* __restrict__ Sr,
                           const unsigned* __restrict__ maxh,
                           const unsigned* __restrict__ maxt,
                           float* __restrict__ sumh,
                           float* __restrict__ sumt, int E) {
    int i = blockIdx.x * blockDim.x + threadIdx.x;
    if (i >= E * 8) return;
    const int e = i >> 3, h = i & 7;
    const int hi = h_idx[e], ti = t_idx[e], r = rel[e];
    const float er = Sr[(size_t)r * 16 + h];
    const float sh = lrelu(Se[(size_t)hi * 16 + h] + er);
    const float st = lrelu(Se[(size_t)ti * 16 + 8 + h] + er);
    atomicAdd(&sumh[(size_t)hi * 8 + h], __expf(sh - fdecode(maxh[(size_t)hi * 8 + h])));
    atomicAdd(&sumt[(size_t)ti * 8 + h], __expf(st - fdecode(maxt[(size_t)ti * 8 + h])));
}

// ---------------------------------------------------------------------------
// Kernel 4: fused alpha + message scatter. One wave32 per edge.
//   a = mean_h( exp(s - max) / (sum + 1e-16) ), then
//   out[hi, 0:128]   += x_r[rel] * a_h   (atomic, L2 resident)
//   out[ti, 128:256] += x_r[rel] * a_t
// ---------------------------------------------------------------------------
__global__ void k_edge_scatter(const int* __restrict__ h_idx,
                               const int* __restrict__ t_idx,
                               const int* __restrict__ rel,
                               const float* __restrict__ Se,
                               const float* __restrict__ Sr,
                               const unsigned* __restrict__ maxh,
                               const unsigned* __restrict__ maxt,
                               const float* __restrict__ sumh,
                               const float* __restrict__ sumt,
                               const float* __restrict__ x_r,
                               float* __restrict__ out, int E) {
    const int wid  = (blockIdx.x * blockDim.x + threadIdx.x) >> 5;
    const int lane = threadIdx.x & 31;
    if (wid >= E) return;  // wave-uniform

    const int hi = h_idx[wid], ti = t_idx[wid], r = rel[wid];

    float ah = 0.0f, at = 0.0f;
#pragma unroll
    for (int h = 0; h < 8; ++h) {
        const float er = Sr[(size_t)r * 16 + h];
        const float sh = lrelu(Se[(size_t)hi * 16 + h] + er);
        const float st = lrelu(Se[(size_t)ti * 16 + 8 + h] + er);
        const float eh = __expf(sh - fdecode(maxh[(size_t)hi * 8 + h]));
        const float et = __expf(st - fdecode(maxt[(size_t)ti * 8 + h]));
        ah += eh / (sumh[(size_t)hi * 8 + h] + 1e-16f);
        at += et / (sumt[(size_t)ti * 8 + h] + 1e-16f);
    }
    ah *= 0.125f;
    at *= 0.125f;

    const float* __restrict__ msg = x_r + (size_t)r * 128;
    float* __restrict__ oh = out + (size_t)hi * 256;
    float* __restrict__ ot = out + (size_t)ti * 256 + 128;
#pragma unroll
    for (int d = lane; d < 128; d += 32) {
        const float mv = msg[d];
        atomicAdd(&oh[d], mv * ah);
        atomicAdd(&ot[d], mv * at);
    }
}

// ---------------------------------------------------------------------------
// Host-side launcher.
// Inputs (setup_inputs order): x_e, x_r, Wh, Wt, Wr, edge_index, rel
// ---------------------------------------------------------------------------
extern "C" void kernel_launch(void* const* d_in, const int* in_sizes, int n_in,
                              void* d_out, int out_size, void* d_ws, size_t ws_size,
                              hipStream_t stream) {
    const float* x_e  = (const float*)d_in[0];
    const float* x_r  = (const float*)d_in[1];
    const float* Wh   = (const float*)d_in[2];
    const float* Wt   = (const float*)d_in[3];
    const float* Wr   = (const float*)d_in[4];
    const int*   eidx = (const int*)d_in[5];
    const int*   rel  = (const int*)d_in[6];

    const int D   = 128;
    const int N_e = in_sizes[0] / D;
    const int N_r = in_sizes[1] / D;
    const int E   = in_sizes[6];

    const int* h_idx = eidx;
    const int* t_idx = eidx + E;

    // Workspace carve-up (256B aligned slices).
    auto align256 = [](size_t x) { return (x + 255) & ~(size_t)255; };
    char* ws = (char*)d_ws;
    float*    Se   = (float*)ws;    ws += align256((size_t)N_e * 16 * sizeof(float));
    float*    Sr   = (float*)ws;    ws += align256((size_t)N_r * 16 * sizeof(float));
    unsigned* maxh = (unsigned*)ws; ws += align256((size_t)N_e * 8 * sizeof(unsigned));
    unsigned* maxt = (unsigned*)ws; ws += align256((size_t)N_e * 8 * sizeof(unsigned));
    float*    sumh = (float*)ws;    ws += align256((size_t)N_e * 8 * sizeof(float));
    float*    sumt = (float*)ws;    ws += align256((size_t)N_e * 8 * sizeof(float));

    float* out = (float*)d_out;

    const int BLK = 256;  // 8 waves per block on wave32

    // 0) init output + softmax stats
    {
        const int n_out = N_e * 256;
        const int n_seg = N_e * 8;
        const int grid  = (n_out + BLK - 1) / BLK;
        k_init<<<grid, BLK, 0, stream>>>(out, sumh, sumt, maxh, maxt, n_out, n_seg);
    }

    // 1) WMMA projections: Se = x_e @ [Wh;Wt]^T  (N=16),
    //    Sr = x_r @ [Wr;Wr]^T (cols 8..15 are a never-read duplicate)
    {
        const int waves   = (N_e + 15) / 16;
        const int threads = waves * 32;
        const int grid    = (threads + BLK - 1) / BLK;
        k_proj_wmma<<<grid, BLK, 0, stream>>>(x_e, Wh, Wt, Se, N_e);
    }
    {
        const int waves   = (N_r + 15) / 16;
        const int threads = waves * 32;
        const int grid    = (threads + BLK - 1) / BLK;
        k_proj_wmma<<<grid, BLK, 0, stream>>>(x_r, Wr, Wr, Sr, N_r);
    }

    // 2) segment max
    {
        const int n    = E * 8;
        const int grid = (n + BLK - 1) / BLK;
        k_edge_max<<<grid, BLK, 0, stream>>>(h_idx, t_idx, rel, Se, Sr, maxh, maxt, E);
    }

    // 3) segment sum of exponentials
    {
        const int n    = E * 8;
        const int grid = (n + BLK - 1) / BLK;
        k_edge_sum<<<grid, BLK, 0, stream>>>(h_idx, t_idx, rel, Se, Sr, maxh, maxt,
                                             sumh, sumt, E);
    }

    // 4) fused alpha + scatter (one wave per edge)
    {
        const long long threads = (long long)E * 32;
        const int grid = (int)((threads + BLK - 1) / BLK);
        k_edge_scatter<<<grid, BLK, 0, stream>>>(h_idx, t_idx, rel, Se, Sr, maxh, maxt,
                                                 sumh, sumt, x_r, out, E);
    }
}